// SRNN_Softmax_67087389164101
// MI455X (gfx1250) — compile-verified
//
#include <hip/hip_runtime.h>
#include <hip/hip_bf16.h>

typedef float v2f __attribute__((ext_vector_type(2)));
typedef float v8f __attribute__((ext_vector_type(8)));

#define HSZ 4096
#define VSZ 4
#define MS  104
#define MD  3

__device__ __forceinline__ float sigmoidf_(float x) {
    return 1.0f / (1.0f + __expf(-x));
}

// ---------------- Kernel 1: h_bar = w_sh @ stack[0] + b_sh + hidden0 ----------------
__global__ void k_hbar(const float* __restrict__ hidden0,
                       const float* __restrict__ stack,
                       const float* __restrict__ w_sh,
                       const float* __restrict__ b_sh,
                       float* __restrict__ hbar) {
    int i = blockIdx.x * blockDim.x + threadIdx.x;
    if (i < HSZ) {
        float s0 = stack[0], s1 = stack[1], s2 = stack[2];
        const float* w = w_sh + (size_t)i * 3;
        float v = b_sh[i] + hidden0[i];
        v = fmaf(w[0], s0, v);
        v = fmaf(w[1], s1, v);
        v = fmaf(w[2], s2, v);
        hbar[i] = v;
    }
}

// ---------------- Kernel 2: gh = w_hh @ h_bar via WMMA, fused GRU gates ----------------
// grid = HSZ/16 workgroups; block = 384 threads = 12 waves:
//   wave w: gate t = w>>2 (r,z,n), k-split s = w&3 (columns [s*1024, s*1024+1024))
// Each wave computes 16 output rows of its gate over its k-split with
// V_WMMA_F32_16X16X4_F32 (A = broadcast h_bar chunk, B = 16x4 weight tile).
__global__ __launch_bounds__(384) void k_gru(
    const float* __restrict__ w_hh, const float* __restrict__ b_hh,
    const float* __restrict__ w_ih, const float* __restrict__ b_ih,
    const float* __restrict__ emb,  const int*   __restrict__ inp,
    const float* __restrict__ hbar,
    float* __restrict__ h_out, float* __restrict__ out_hidden) {

    __shared__ float sh_hbar[HSZ];
    __shared__ float sh_gh[48];

    const int tid = threadIdx.x;
    for (int i = tid; i < HSZ; i += 384) sh_hbar[i] = hbar[i];
    if (tid < 48) sh_gh[tid] = 0.0f;
    __syncthreads();

    const int lane = tid & 31;
    const int wv   = tid >> 5;       // 0..11
    const int t    = wv >> 2;        // gate: 0=r,1=z,2=n
    const int s    = wv & 3;         // k-split
    const int m    = lane & 15;      // row within 16-row tile / B column
    const int koff = (lane >> 4) * 2;

    const int row   = t * HSZ + (blockIdx.x << 4) + m;
    const int kbase = s * (HSZ / 4) + koff;     // 1024-wide split

    const float* wp = w_hh + (size_t)row * HSZ + kbase;   // weight row walk
    const float* ap = sh_hbar + kbase;                    // LDS h_bar walk

    v8f c0 = {}, c1 = {}, c2 = {}, c3 = {};
    for (int i = 0; i < 1024; i += 16) {
        v2f a0 = *(const v2f*)(ap + i);
        v2f b0 = *(const v2f*)(wp + i);
        c0 = __builtin_amdgcn_wmma_f32_16x16x4_f32(false, a0, false, b0,
                                                   (short)0, c0, false, false);
        v2f a1 = *(const v2f*)(ap + i + 4);
        v2f b1 = *(const v2f*)(wp + i + 4);
        c1 = __builtin_amdgcn_wmma_f32_16x16x4_f32(false, a1, false, b1,
                                                   (short)0, c1, false, false);
        v2f a2 = *(const v2f*)(ap + i + 8);
        v2f b2 = *(const v2f*)(wp + i + 8);
        c2 = __builtin_amdgcn_wmma_f32_16x16x4_f32(false, a2, false, b2,
                                                   (short)0, c2, false, false);
        v2f a3 = *(const v2f*)(ap + i + 12);
        v2f b3 = *(const v2f*)(wp + i + 12);
        c3 = __builtin_amdgcn_wmma_f32_16x16x4_f32(false, a3, false, b3,
                                                   (short)0, c3, false, false);
    }

    // D row M=0 lives in accumulator VGPR0, lanes 0..15 = output rows n=0..15
    float y = c0[0] + c1[0] + c2[0] + c3[0];
    if (lane < 16) atomicAdd(&sh_gh[t * 16 + lane], y);   // ds_add_f32 across k-splits
    __syncthreads();

    if (tid < 16) {
        const int j = (blockIdx.x << 4) + tid;
        float ghr = sh_gh[tid]      + b_hh[j];
        float ghz = sh_gh[16 + tid] + b_hh[HSZ + j];
        float ghn = sh_gh[32 + tid] + b_hh[2 * HSZ + j];

        const int e = inp[0];
        const float x0 = emb[(size_t)e * VSZ + 0];
        const float x1 = emb[(size_t)e * VSZ + 1];
        const float x2 = emb[(size_t)e * VSZ + 2];
        const float x3 = emb[(size_t)e * VSZ + 3];

        const float* wi;
        wi = w_ih + (size_t)j * VSZ;
        float gir = fmaf(wi[0], x0, fmaf(wi[1], x1, fmaf(wi[2], x2, fmaf(wi[3], x3, b_ih[j]))));
        wi = w_ih + (size_t)(HSZ + j) * VSZ;
        float giz = fmaf(wi[0], x0, fmaf(wi[1], x1, fmaf(wi[2], x2, fmaf(wi[3], x3, b_ih[HSZ + j]))));
        wi = w_ih + (size_t)(2 * HSZ + j) * VSZ;
        float gin = fmaf(wi[0], x0, fmaf(wi[1], x1, fmaf(wi[2], x2, fmaf(wi[3], x3, b_ih[2 * HSZ + j]))));

        float r  = sigmoidf_(gir + ghr);
        float z  = sigmoidf_(giz + ghz);
        float n  = tanhf(gin + r * ghn);
        float hb = sh_hbar[j];
        float hv = (1.0f - z) * n + z * hb;

        h_out[j]      = hv;   // workspace copy for k_heads
        out_hidden[j] = hv;   // hidden output slot
    }
}

// ---------------- Kernel 3: heads + softmax + stack blend ----------------
__global__ __launch_bounds__(256) void k_heads(
    const float* __restrict__ h,
    const float* __restrict__ w_y, const float* __restrict__ b_y,
    const float* __restrict__ w_a, const float* __restrict__ b_a,
    const float* __restrict__ w_n, const float* __restrict__ b_n,
    const float* __restrict__ stack, float* __restrict__ out) {

    __shared__ float sh_dot[9];
    const int tid  = threadIdx.x;
    const int lane = tid & 31;
    const int wv   = tid >> 5;   // 8 waves

    for (int r = wv; r < 9; r += 8) {
        const float* p = (r < 4) ? (w_y + (size_t)r * HSZ)
                       : (r < 6) ? (w_a + (size_t)(r - 4) * HSZ)
                                 : (w_n + (size_t)(r - 6) * HSZ);
        float acc = 0.0f;
        for (int k = lane; k < HSZ; k += 32) acc = fmaf(p[k], h[k], acc);
        for (int off = 16; off > 0; off >>= 1) acc += __shfl_down(acc, off, 32);
        if (lane == 0) sh_dot[r] = acc;
    }
    __syncthreads();

    if (tid < 4) out[tid] = sigmoidf_(sh_dot[tid] + b_y[tid]);

    // 2-way softmax + new element (computed redundantly per thread; trivial)
    const float v0 = sh_dot[4] + b_a[0];
    const float v1 = sh_dot[5] + b_a[1];
    const float mx = fmaxf(v0, v1);
    const float e0 = __expf(v0 - mx), e1 = __expf(v1 - mx);
    const float inv = 1.0f / (e0 + e1);
    const float a0 = e0 * inv, a1 = e1 * inv;
    const float ne0 = sigmoidf_(sh_dot[6] + b_n[0]);
    const float ne1 = sigmoidf_(sh_dot[7] + b_n[1]);
    const float ne2 = sigmoidf_(sh_dot[8] + b_n[2]);

    for (int idx = tid; idx < MS * MD; idx += 256) {
        const int i = idx / MD;
        const int d = idx - i * MD;
        const float ne   = (d == 0) ? ne0 : ((d == 1) ? ne1 : ne2);
        const float push = (i == 0) ? ne : stack[(size_t)(i - 1) * MD + d];
        const float pop  = (i == MS - 1) ? 0.0f : stack[(size_t)(i + 1) * MD + d];
        out[4 + HSZ + idx] = a0 * push + a1 * pop;
    }
}

// ---------------- launch ----------------
extern "C" void kernel_launch(void* const* d_in, const int* in_sizes, int n_in,
                              void* d_out, int out_size, void* d_ws, size_t ws_size,
                              hipStream_t stream) {
    const int*   inp     = (const int*)  d_in[0];
    const float* hidden0 = (const float*)d_in[1];
    const float* stack   = (const float*)d_in[2];
    const float* emb     = (const float*)d_in[3];
    const float* w_ih    = (const float*)d_in[4];
    const float* w_hh    = (const float*)d_in[5];
    const float* b_ih    = (const float*)d_in[6];
    const float* b_hh    = (const float*)d_in[7];
    const float* w_y     = (const float*)d_in[8];
    const float* b_y     = (const float*)d_in[9];
    const float* w_n     = (const float*)d_in[10];
    const float* b_n     = (const float*)d_in[11];
    const float* w_a     = (const float*)d_in[12];
    const float* b_a     = (const float*)d_in[13];
    const float* w_sh    = (const float*)d_in[14];
    const float* b_sh    = (const float*)d_in[15];

    float* out  = (float*)d_out;
    float* hbar = (float*)d_ws;       // HSZ floats
    float* hvec = hbar + HSZ;         // HSZ floats

    k_hbar<<<(HSZ + 255) / 256, 256, 0, stream>>>(hidden0, stack, w_sh, b_sh, hbar);
    k_gru<<<HSZ / 16, 384, 0, stream>>>(w_hh, b_hh, w_ih, b_ih, emb, inp,
                                        hbar, hvec, out + 4);
    k_heads<<<1, 256, 0, stream>>>(hvec, w_y, b_y, w_a, b_a, w_n, b_n, stack, out);
}